// ImplicitConv_68169720922894
// MI455X (gfx1250) — compile-verified
//
#include <hip/hip_runtime.h>

// ===========================================================================
// Spectral Laplacian^{-1} filter (SMOOTHNESS=2, SCALE=1) on 32 x 1024 x 1024.
//
//  y = Re{ (C+iS) [ lam .* ((C-iS) X (C-iS)) ] (C+iS) }
//  with C[i][j]=cos(2*pi*i*j/N), S[i][j]=sin(2*pi*i*j/N)  (symmetric),
//  lam(k1,k2) = SCALE / ((4-2cos(2pi k1/N)-2cos(2pi k2/N)) * N*N), lam(0,0)=0.
//
//  Every stage is a real f32 GEMM -> V_WMMA_F32_16X16X4_F32, with K-panels
//  double-buffered in LDS and filled by GLOBAL_LOAD_ASYNC_TO_LDS_B128.
//  All +/-1 stage signs are absorbed into precomputed -C / -S matrices so the
//  inner loop is pure ds_load -> v_wmma (no VALU scaling on the operand path).
// ===========================================================================

typedef float v2f __attribute__((ext_vector_type(2)));
typedef float v4f __attribute__((ext_vector_type(4)));
typedef float v8f __attribute__((ext_vector_type(8)));
typedef int   v4i __attribute__((ext_vector_type(4)));

#define NSIDE 1024
#define NIMG  32
#define CHUNK 8                     // images per batched pass
#define KB_CNT (NSIDE / 16)         // 64 K-panels of width 16
#define TWO_PI 6.283185307179586f

#if __has_builtin(__builtin_amdgcn_global_load_async_to_lds_b128)
#define HAS_ASYNC_LDS 1
#else
#define HAS_ASYNC_LDS 0
#endif

typedef __attribute__((address_space(3))) v4i las_v4i;   // LDS-side b128 payload

__device__ __forceinline__ void wait_async_zero() {
#if HAS_ASYNC_LDS
#if __has_builtin(__builtin_amdgcn_s_wait_asynccnt)
    __builtin_amdgcn_s_wait_asynccnt(0);
#else
    asm volatile("s_wait_asynccnt 0x0" ::: "memory");
#endif
#endif
}

// --------------------------------------------------------------------------
// Build DFT cosine/sine matrices and their negations:
//   C[i][j]=cos(2pi*(i*j mod N)/N), S=sin, Cn=-C, Sn=-S.
// --------------------------------------------------------------------------
__global__ void build_trig(float* __restrict__ Cm, float* __restrict__ Sm,
                           float* __restrict__ Cn, float* __restrict__ Sn) {
    int j = blockIdx.x * blockDim.x + threadIdx.x;   // column
    int i = blockIdx.y;                              // row
    unsigned p = ((unsigned)i * (unsigned)j) & (NSIDE - 1);
    float ang = (float)p * (TWO_PI / (float)NSIDE);
    float s, c;
    __sincosf(ang, &s, &c);
    long idx = (long)i * NSIDE + j;
    Cm[idx] = c;
    Sm[idx] = s;
    Cn[idx] = -c;
    Sn[idx] = -s;
}

// --------------------------------------------------------------------------
// lam(k1,k2) = SCALE / (|Bh| * N^2), DC zeroed.  |Bh| = 4-2cos(a)-2cos(b).
// --------------------------------------------------------------------------
__global__ void build_lam(float* __restrict__ lam) {
    int j = blockIdx.x * blockDim.x + threadIdx.x;
    int i = blockIdx.y;
    float a = 4.0f - 2.0f * __cosf((float)i * (TWO_PI / (float)NSIDE))
                   - 2.0f * __cosf((float)j * (TWO_PI / (float)NSIDE));
    const float inv_n2 = 1.0f / ((float)NSIDE * (float)NSIDE);
    float v = (i == 0 && j == 0) ? 0.0f : (1.0f /*SCALE*/ * inv_n2) / a;
    lam[(long)i * NSIDE + j] = v;
}

// --------------------------------------------------------------------------
// Elementwise spectral scaling: Re *= lam, Im *= lam (batched).
// --------------------------------------------------------------------------
__global__ void apply_lam(float* __restrict__ Re, float* __restrict__ Im,
                          const float* __restrict__ lam) {
    long i = (long)blockIdx.x * blockDim.x + threadIdx.x;   // 0 .. N*N-1
    long b = blockIdx.y;
    const long M = (long)NSIDE * NSIDE;
    float f = lam[i];
    Re[b * M + i] *= f;
    Im[b * M + i] *= f;
}

// --------------------------------------------------------------------------
// Stage one 128x16 A-panel and one 16x128 B-panel into an LDS buffer.
// Async path: GLOBAL_LOAD_ASYNC_TO_LDS_B128 (16B/lane, ASYNCcnt-tracked).
// Fallback: VGPR float4 staging (ds_store), same addressing.
// --------------------------------------------------------------------------
__device__ __forceinline__ void stage_panels(const float* __restrict__ A,
                                             const float* __restrict__ B,
                                             int kb, int gm0, int gn0, int tid,
                                             float* __restrict__ Albuf,
                                             float* __restrict__ Blbuf) {
#pragma unroll
    for (int i = 0; i < 2; ++i) {
        int t = tid * 2 + i;           // 0..511 float4 slots
        int e = t * 4;                 // float index within panel
        int r  = e >> 4,  c  = e & 15;     // A: [row][k], 16 wide
        int kk = e >> 7,  cc = e & 127;    // B: [k][col], 128 wide
        const float* ga = &A[(long)(gm0 + r) * NSIDE + kb * 16 + c];
        const float* gb = &B[(long)(kb * 16 + kk) * NSIDE + gn0 + cc];
#if HAS_ASYNC_LDS
        __builtin_amdgcn_global_load_async_to_lds_b128(
            (v4i*)(void*)ga, (las_v4i*)&Albuf[e], 0, 0);
        __builtin_amdgcn_global_load_async_to_lds_b128(
            (v4i*)(void*)gb, (las_v4i*)&Blbuf[e], 0, 0);
#else
        *(v4f*)&Albuf[e] = *(const v4f*)ga;
        *(v4f*)&Blbuf[e] = *(const v4f*)gb;
#endif
    }
}

// --------------------------------------------------------------------------
// Batched two-pass GEMM:  D[b] = A1[b]*B1[b] + A2[b]*B2[b]
// (strides 0 for shared/broadcast operands; signs pre-baked into operands).
// All matrices 1024x1024 f32.
//
// WG tile 128x128, 8 wave32s in a 2x4 arrangement, each wave owns 64x32
// (4x2 tiles of 16x16).  Inner product via V_WMMA_F32_16X16X4_F32.
// Double-buffered LDS K-panels; panel kb+1 streams in asynchronously while
// WMMAs consume panel kb; one barrier per K-iteration.
// --------------------------------------------------------------------------
__launch_bounds__(256)
__global__ void gemm_dft(const float* __restrict__ A1, long sA1,
                         const float* __restrict__ B1, long sB1,
                         const float* __restrict__ A2, long sA2,
                         const float* __restrict__ B2, long sB2,
                         float* __restrict__ D, long sD, int npass)
{
    __shared__ float Alds[2][128 * 16];   // [buf][row][k]   16 wide
    __shared__ float Blds[2][16 * 128];   // [buf][k][col]   128 wide

    const int tid   = threadIdx.x;        // 0..255
    const int lane  = tid & 31;
    const int wave  = tid >> 5;           // 0..7
    const int m0w   = (wave >> 2) * 64;   // wave row origin inside WG tile
    const int n0w   = (wave & 3) * 32;    // wave col origin inside WG tile
    const int lhalf = lane >> 4;          // 0 | 1  (K sub-pair select)
    const int l15   = lane & 15;

    const int gm0 = blockIdx.y * 128;
    const int gn0 = blockIdx.x * 128;
    const int b   = blockIdx.z;

    const float* Ab[2] = { A1 + (long)b * sA1,
                           (npass > 1 ? A2 + (long)b * sA2 : A1) };
    const float* Bb[2] = { B1 + (long)b * sB1,
                           (npass > 1 ? B2 + (long)b * sB2 : B1) };

    v8f acc[4][2];
#pragma unroll
    for (int mt = 0; mt < 4; ++mt)
#pragma unroll
        for (int nt = 0; nt < 2; ++nt)
            acc[mt][nt] = (v8f){0.f, 0.f, 0.f, 0.f, 0.f, 0.f, 0.f, 0.f};

    const int T = npass * KB_CNT;         // flattened (pass, kb) iterations

    // ---- prologue: stage panel 0 into buffer 0 ----
    stage_panels(Ab[0], Bb[0], 0, gm0, gn0, tid, Alds[0], Blds[0]);
    wait_async_zero();
    __syncthreads();

    int cur = 0;
    for (int t = 0; t < T; ++t) {
        // ---- kick off DMA for the next K-panel into the free buffer ----
        if (t + 1 < T) {
            int pn = (t + 1) >> 6;            // / KB_CNT
            int kn = (t + 1) & (KB_CNT - 1);
            stage_panels(Ab[pn], Bb[pn], kn, gm0, gn0, tid,
                         Alds[cur ^ 1], Blds[cur ^ 1]);
        }

        const float* __restrict__ Al = Alds[cur];
        const float* __restrict__ Bl = Blds[cur];

        // ---- 4 WMMA K-steps of 4 over the 16-wide panel ----
#pragma unroll
        for (int k0 = 0; k0 < 16; k0 += 4) {
            const int kk = k0 + 2 * lhalf;   // lanes 0-15 -> K{k0,k0+1}; 16-31 -> K{k0+2,k0+3}

            v2f af[4];
#pragma unroll
            for (int mt = 0; mt < 4; ++mt) {
                int r = m0w + mt * 16 + l15;
                af[mt].x = Al[r * 16 + kk];
                af[mt].y = Al[r * 16 + kk + 1];
            }
            v2f bf[2];
#pragma unroll
            for (int nt = 0; nt < 2; ++nt) {
                int cc = n0w + nt * 16 + l15;
                bf[nt].x = Bl[kk * 128 + cc];
                bf[nt].y = Bl[(kk + 1) * 128 + cc];
            }
#pragma unroll
            for (int mt = 0; mt < 4; ++mt)
#pragma unroll
                for (int nt = 0; nt < 2; ++nt)
                    acc[mt][nt] = __builtin_amdgcn_wmma_f32_16x16x4_f32(
                        false, af[mt], false, bf[nt],
                        (short)0, acc[mt][nt], false, false);
        }

        // ---- retire this wave's DMAs, then WG-wide handoff ----
        wait_async_zero();
        __syncthreads();
        cur ^= 1;
    }

    // ---- epilogue: C/D layout -> VGPR v holds row (v + 8*lhalf), col l15 ----
    float* Dp = D + (long)b * sD;
#pragma unroll
    for (int mt = 0; mt < 4; ++mt)
#pragma unroll
        for (int nt = 0; nt < 2; ++nt)
#pragma unroll
            for (int v = 0; v < 8; ++v) {
                int row = gm0 + m0w + mt * 16 + v + 8 * lhalf;
                int col = gn0 + n0w + nt * 16 + l15;
                Dp[(long)row * NSIDE + col] = acc[mt][nt][v];
            }
}

// ===========================================================================
// Host-side orchestration (stream-only; graph-capture safe).
// ===========================================================================
extern "C" void kernel_launch(void* const* d_in, const int* in_sizes, int n_in,
                              void* d_out, int out_size, void* d_ws, size_t ws_size,
                              hipStream_t stream) {
    const float* x = (const float*)d_in[0];
    // d_in[1] (K, 3x3) is the fixed 5-point Laplacian; its spectrum is built
    // analytically in build_lam, so it is not consumed on-device.

    const long M = (long)NSIDE * NSIDE;        // 1M elements per matrix
    float* ws  = (float*)d_ws;
    float* Cm  = ws;                            // 1M
    float* Sm  = Cm + M;                        // 1M
    float* Cn  = Sm + M;                        // 1M  (-C)
    float* Sn  = Cn + M;                        // 1M  (-S)
    float* lam = Sn + M;                        // 1M
    float* U   = lam + M;                       // CHUNK * 1M
    float* V   = U + (long)CHUNK * M;
    float* Re  = V + (long)CHUNK * M;
    float* Im  = Re + (long)CHUNK * M;

    dim3 blk(256);
    dim3 gInit(NSIDE / 256, NSIDE);
    build_trig<<<gInit, blk, 0, stream>>>(Cm, Sm, Cn, Sn);
    build_lam <<<gInit, blk, 0, stream>>>(lam);

    dim3 g(NSIDE / 128, NSIDE / 128, CHUNK);    // 8 x 8 x 8
    dim3 gLam((unsigned)(M / 256), CHUNK);

    for (int c = 0; c < NIMG / CHUNK; ++c) {
        const float* xc = x + (long)c * CHUNK * M;
        float*       oc = (float*)d_out + (long)c * CHUNK * M;

        // Forward row transform:  U = X*C ; V = X*S
        gemm_dft<<<g, blk, 0, stream>>>(xc, M, Cm, 0,
                                        nullptr, 0, nullptr, 0, U, M, 1);
        gemm_dft<<<g, blk, 0, stream>>>(xc, M, Sm, 0,
                                        nullptr, 0, nullptr, 0, V, M, 1);
        // Forward col transform:  Re = C*U + (-S)*V ; Im = (-C)*V + (-S)*U
        gemm_dft<<<g, blk, 0, stream>>>(Cm, 0, U, M,
                                        Sn, 0, V, M, Re, M, 2);
        gemm_dft<<<g, blk, 0, stream>>>(Cn, 0, V, M,
                                        Sn, 0, U, M, Im, M, 2);
        // Spectral power-law filter
        apply_lam<<<gLam, blk, 0, stream>>>(Re, Im, lam);
        // Inverse col transform:  U = C*Re + (-S)*Im ; V = C*Im + S*Re
        gemm_dft<<<g, blk, 0, stream>>>(Cm, 0, Re, M,
                                        Sn, 0, Im, M, U, M, 2);
        gemm_dft<<<g, blk, 0, stream>>>(Cm, 0, Im, M,
                                        Sm, 0, Re, M, V, M, 2);
        // Inverse row transform (real part):  out = U*C + V*(-S)
        gemm_dft<<<g, blk, 0, stream>>>(U, M, Cm, 0,
                                        V, M, Sn, 0, oc, M, 2);
    }
}